// GraphAttentionLayer_32495722562235
// MI455X (gfx1250) — compile-verified
//
#include <hip/hip_runtime.h>

typedef __attribute__((ext_vector_type(2))) float v2f;
typedef __attribute__((ext_vector_type(8))) float v8f;

#define ALPHA    0.2f
#define NEG_BIG  -9000000000000000.0f
#define N_NODES  2048
#define FDIM     128
#define NBATCH   4

// ---------------------------------------------------------------------------
// Kernel 1: Wh = h @ W   (f32 WMMA 16x16x4)
// Block: 256 threads (8 waves) handles one 16-row tile x all 128 output cols.
// ---------------------------------------------------------------------------
__global__ __launch_bounds__(256) void gat_wh_kernel(const float* __restrict__ h,
                                                     const float* __restrict__ W,
                                                     float* __restrict__ Wh) {
    __shared__ float sh[16 * FDIM];   // 8KB h tile
    const int tile = blockIdx.x;                       // 0 .. B*N/16-1
    const int b    = tile / (N_NODES / 16);
    const int i0   = (tile % (N_NODES / 16)) * 16;
    const int t    = threadIdx.x;

    const float* hbase = h + ((size_t)b * N_NODES + i0) * FDIM;
    for (int k = t; k < 16 * FDIM; k += 256) sh[k] = hbase[k];
    __syncthreads();

    const int wave  = t >> 5;          // 0..7 -> Fout tile
    const int lane  = t & 31;
    const int row   = lane & 15;       // M index for A, N index for B/D
    const int kslot = lane >> 4;       // 0: K={0,1}, 1: K={2,3}
    const int n0    = wave * 16;

    v8f acc = {};
    for (int k0 = 0; k0 < FDIM; k0 += 4) {
        v2f A, B;
        A.x = sh[row * FDIM + k0 + kslot * 2];
        A.y = sh[row * FDIM + k0 + kslot * 2 + 1];
        B.x = W[(size_t)(k0 + kslot * 2)     * FDIM + n0 + row];
        B.y = W[(size_t)(k0 + kslot * 2 + 1) * FDIM + n0 + row];
        acc = __builtin_amdgcn_wmma_f32_16x16x4_f32(false, A, false, B,
                                                    (short)0, acc, false, false);
    }

    float* out = Wh + ((size_t)b * N_NODES + i0) * FDIM + n0 + row;
#pragma unroll
    for (int v = 0; v < 8; ++v) {
        int orow = v + 8 * kslot;                     // D layout: M = v + 8*(lane/16)
        out[(size_t)orow * FDIM] = acc[v];
    }
}

// ---------------------------------------------------------------------------
// Kernel 2: f1 = Wh@a1, f2 = Wh@a2   (one wave32 per row)
// ---------------------------------------------------------------------------
__global__ __launch_bounds__(256) void gat_f_kernel(const float* __restrict__ Wh,
                                                    const float* __restrict__ a,
                                                    float* __restrict__ f1,
                                                    float* __restrict__ f2) {
    const int row  = blockIdx.x * 8 + (threadIdx.x >> 5);   // 0 .. B*N-1
    const int lane = threadIdx.x & 31;
    const float* wr = Wh + (size_t)row * FDIM;
    float s1 = 0.f, s2 = 0.f;
#pragma unroll
    for (int c = 0; c < FDIM; c += 32) {
        float v = wr[c + lane];
        s1 += v * a[c + lane];
        s2 += v * a[FDIM + c + lane];
    }
#pragma unroll
    for (int m = 16; m >= 1; m >>= 1) {
        s1 += __shfl_xor(s1, m, 32);
        s2 += __shfl_xor(s2, m, 32);
    }
    if (lane == 0) { f1[row] = s1; f2[row] = s2; }
}

// ---------------------------------------------------------------------------
// Kernel 3: fused masked-softmax + P @ Wh  (two-pass, f32 WMMA 16x16x4)
// Block: 256 threads = (batch b, 16 query rows i0..i0+15).
//   Phase 1: exact per-row max and exp-sum.
//   Phase 2: wave w accumulates its j-slice [w*256, w*256+256) into a 16x128
//            register tile (8 WMMA accumulators), combined via LDS f32 atomics.
// ---------------------------------------------------------------------------
__global__ __launch_bounds__(256) void gat_attn_kernel(const float* __restrict__ Wh,
                                                       const int*   __restrict__ adj,
                                                       const float* __restrict__ f1,
                                                       const float* __restrict__ f2,
                                                       float* __restrict__ out) {
    __shared__ float s_f2[N_NODES];       // 8KB
    __shared__ float s_acc[16 * FDIM];    // 8KB output tile accumulator
    __shared__ float s_m[16];
    __shared__ float s_l[16];

    const int blk  = blockIdx.x;
    const int b    = blk / (N_NODES / 16);
    const int i0   = (blk % (N_NODES / 16)) * 16;
    const int t    = threadIdx.x;
    const int wave = t >> 5;
    const int lane = t & 31;

    for (int j = t; j < N_NODES; j += 256) s_f2[j] = f2[(size_t)b * N_NODES + j];
    for (int k = t; k < 16 * FDIM; k += 256) s_acc[k] = 0.f;
    __syncthreads();

    // ---- Phase 1: row statistics. Wave w: lanes 0-15 -> row 2w, 16-31 -> row 2w+1.
    {
        const int half = lane >> 4;
        const int r    = wave * 2 + half;
        const int l16  = lane & 15;
        const int* arow = adj + (size_t)(i0 + r) * N_NODES;
        const float fr  = f1[(size_t)b * N_NODES + i0 + r];

        float mx = NEG_BIG;
        for (int j = l16; j < N_NODES; j += 16) {       // coalesced per 16-lane group
            float x = fr + s_f2[j];
            float s = (arow[j] > 0) ? (x > 0.f ? x : ALPHA * x) : NEG_BIG;
            mx = fmaxf(mx, s);
        }
#pragma unroll
        for (int m = 8; m >= 1; m >>= 1) mx = fmaxf(mx, __shfl_xor(mx, m, 32));

        float sum = 0.f;
        for (int j = l16; j < N_NODES; j += 16) {
            float x = fr + s_f2[j];
            float s = (arow[j] > 0) ? (x > 0.f ? x : ALPHA * x) : NEG_BIG;
            sum += __expf(s - mx);                       // masked -> exp(-9e15)=0
        }
#pragma unroll
        for (int m = 8; m >= 1; m >>= 1) sum += __shfl_xor(sum, m, 32);

        if (l16 == 0) { s_m[r] = mx; s_l[r] = sum; }
    }
    __syncthreads();

    // ---- Phase 2: WMMA accumulation over this wave's j slice.
    const int ln16  = lane & 15;          // M index of A, N index of B/D
    const int kslot = lane >> 4;          // K pair selector
    const float rm  = s_m[ln16];
    const float rf1 = f1[(size_t)b * N_NODES + i0 + ln16];
    const int* arow = adj + (size_t)(i0 + ln16) * N_NODES;
    const float* whb = Wh + (size_t)b * N_NODES * FDIM;

    v8f acc[8] = {};                      // 8 Fout tiles x 8 VGPRs = 16x128 partial
    const int jend = wave * 256 + 256;
    for (int j0 = wave * 256; j0 < jend; j0 += 4) {
        const int jA = j0 + kslot * 2;
        float x0 = rf1 + s_f2[jA];
        float s0 = (arow[jA] > 0) ? (x0 > 0.f ? x0 : ALPHA * x0) : NEG_BIG;
        float x1 = rf1 + s_f2[jA + 1];
        float s1 = (arow[jA + 1] > 0) ? (x1 > 0.f ? x1 : ALPHA * x1) : NEG_BIG;
        v2f A;
        A.x = __expf(s0 - rm);
        A.y = __expf(s1 - rm);

        const float* w0 = whb + (size_t)jA * FDIM + ln16;
        const float* w1 = w0 + FDIM;
#pragma unroll
        for (int n = 0; n < 8; ++n) {
            v2f B;
            B.x = w0[n * 16];
            B.y = w1[n * 16];
            acc[n] = __builtin_amdgcn_wmma_f32_16x16x4_f32(false, A, false, B,
                                                           (short)0, acc[n], false, false);
        }
    }

    // Combine the 8 waves' partial tiles via LDS float atomics (ds_add_f32).
#pragma unroll
    for (int n = 0; n < 8; ++n) {
#pragma unroll
        for (int v = 0; v < 8; ++v) {
            int orow = v + 8 * kslot;
            atomicAdd(&s_acc[orow * FDIM + n * 16 + ln16], acc[n][v]);
        }
    }
    __syncthreads();

    // Normalize by row sums and write out.
    for (int k = t; k < 16 * FDIM; k += 256) {
        int r = k >> 7;                   // k / FDIM
        out[((size_t)b * N_NODES + i0 + r) * FDIM + (k & (FDIM - 1))] = s_acc[k] / s_l[r];
    }
}

// ---------------------------------------------------------------------------
extern "C" void kernel_launch(void* const* d_in, const int* in_sizes, int n_in,
                              void* d_out, int out_size, void* d_ws, size_t ws_size,
                              hipStream_t stream) {
    const float* h   = (const float*)d_in[0];   // [4,2048,128]
    const int*   adj = (const int*)  d_in[1];   // [2048,2048]
    const float* W   = (const float*)d_in[2];   // [128,128]
    const float* a   = (const float*)d_in[3];   // [256,1]
    float* out = (float*)d_out;                 // [4,2048,128]

    float* Wh = (float*)d_ws;                                   // 4 MB
    float* f1 = Wh + (size_t)NBATCH * N_NODES * FDIM;           // 32 KB
    float* f2 = f1 + (size_t)NBATCH * N_NODES;                  // 32 KB

    gat_wh_kernel  <<<NBATCH * (N_NODES / 16), 256, 0, stream>>>(h, W, Wh);
    gat_f_kernel   <<<NBATCH * (N_NODES / 8),  256, 0, stream>>>(Wh, a, f1, f2);
    gat_attn_kernel<<<NBATCH * (N_NODES / 16), 256, 0, stream>>>(Wh, adj, f1, f2, out);
}